// ImgMeBlock_12266426598094
// MI455X (gfx1250) — compile-verified
//
#include <hip/hip_runtime.h>

// ---------------------------------------------------------------------------
// Token top-K pruning for MI455X (gfx1250, wave32).
//   scores = emb . W + b        (512 MB read)   -> WMMA f32 16x16x4 path
//   topK indices by rank-count  (tiny)          -> LDS staged
//   gather selected rows        (256MB r + 256MB w, b128 vectorized)
// Problem is HBM-bound (~1 GB total @ 23.3 TB/s ~= 45us); softmax skipped
// (monotonic, does not change top-k ordering).
// ---------------------------------------------------------------------------

typedef __attribute__((ext_vector_type(2))) float v2f;
typedef __attribute__((ext_vector_type(8))) float v8f;

#define BB 32
#define NN 4096
#define DD 1024
#define KK 2048

// ---------------------------------------------------------------------------
// Pass 1: scores[b][n] = dot(emb[b][n][:], W) + bias via V_WMMA_F32_16X16X4_F32.
// One wave handles a 16-token tile. A = 16x4 tile of embeddings, B = 4x16 with
// W chunk replicated across all 16 columns, so every column of D is the score.
// A-matrix f32 16x4 layout (ISA 7.12.2): lane = M (0..15 twice),
//   VGPR0: lanes0-15 K=0, lanes16-31 K=2 ; VGPR1: K=1 / K=3.
// B-matrix 4x16 uses the symmetric K striping, and since all N columns are
// identical only the K mapping matters.
// ---------------------------------------------------------------------------
__global__ void tp_score_wmma(const float* __restrict__ emb,
                              const float* __restrict__ W,
                              const float* __restrict__ bias,
                              float* __restrict__ scores) {
  const int wavesPerBlock = blockDim.x >> 5;
  const int waveId        = threadIdx.x >> 5;
  const int lane          = threadIdx.x & 31;

  const int tile          = blockIdx.x * wavesPerBlock + waveId; // 16-token tile
  const int tilesPerBatch = NN >> 4;                             // 256
  const int b             = tile / tilesPerBatch;
  const int row0          = (tile % tilesPerBatch) << 4;

  const int row  = lane & 15;          // M this lane holds
  const int ksel = (lane >> 4) << 1;   // K offset: 0 for lanes 0-15, 2 for 16-31

  const float* __restrict__ rowp = emb + ((size_t)b * NN + row0 + row) * DD + ksel;

  v8f acc = {};
  for (int k0 = 0; k0 < DD; k0 += 4) {
    v2f a, bv;
    a.x  = rowp[k0 + 0];
    a.y  = rowp[k0 + 1];
    bv.x = W[k0 + ksel + 0];
    bv.y = W[k0 + ksel + 1];
    // D = A(16x4) * B(4x16) + C, f32 accumulate
    acc = __builtin_amdgcn_wmma_f32_16x16x4_f32(
        /*neg_a=*/false, a, /*neg_b=*/false, bv,
        /*c_mod=*/(short)0, acc, /*reuse_a=*/false, /*reuse_b=*/false);
  }

  // D layout: VGPR v -> M = v + 8*(lane/16), column = lane%16 (all cols equal).
  // Lanes 0 and 16 write the 16 scores of this tile.
  if ((lane & 15) == 0) {
    const int   half = lane >> 4;
    const float bsc  = bias[0];
    float* __restrict__ sp = scores + (size_t)b * NN + row0 + (half << 3);
#pragma unroll
    for (int v = 0; v < 8; ++v) sp[v] = acc[v] + bsc;
  }
}

// ---------------------------------------------------------------------------
// Pass 2: exact top-K via rank counting (matches jax.lax.top_k ordering:
// descending score, stable on ties by lower index). Each block ranks 256
// elements of one batch row; all 4096 scores staged in LDS (16 KB).
// rank(i) = #{j: s_j > s_i} + #{j < i: s_j == s_i};  rank < K -> out pos = rank.
// ---------------------------------------------------------------------------
__global__ void tp_rank_topk(const float* __restrict__ scores,
                             int* __restrict__ top_idx) {
  __shared__ float s_sc[NN];
  const int blocksPerBatch = NN / 256;                // 16
  const int b   = blockIdx.x / blocksPerBatch;
  const int seg = blockIdx.x % blocksPerBatch;

  const float* __restrict__ row = scores + (size_t)b * NN;
  for (int i = threadIdx.x; i < NN; i += blockDim.x) s_sc[i] = row[i];
  __syncthreads();

  const int   i  = seg * 256 + threadIdx.x;
  const float si = s_sc[i];
  int rank = 0;
#pragma unroll 4
  for (int j = 0; j < NN; ++j) {
    const float sj = s_sc[j];
    rank += (sj > si) | ((sj == si) & (j < i));
  }
  if (rank < KK) top_idx[(size_t)b * KK + rank] = i;
}

// ---------------------------------------------------------------------------
// Pass 3: gather. One block per output row; 256 threads x float4 = 4 KB row.
// Pure b128 streaming: ~256 MB read + 256 MB write at HBM rate.
// ---------------------------------------------------------------------------
__global__ void tp_gather(const float* __restrict__ emb,
                          const int* __restrict__ top_idx,
                          float* __restrict__ out) {
  const int b = blockIdx.x / KK;
  const int k = blockIdx.x % KK;
  const int src = top_idx[(size_t)b * KK + k];

  const float4* __restrict__ s =
      (const float4*)(emb + ((size_t)b * NN + src) * DD);
  float4* __restrict__ d = (float4*)(out + ((size_t)b * KK + k) * DD);
  d[threadIdx.x] = s[threadIdx.x];
}

// ---------------------------------------------------------------------------
extern "C" void kernel_launch(void* const* d_in, const int* in_sizes, int n_in,
                              void* d_out, int out_size, void* d_ws, size_t ws_size,
                              hipStream_t stream) {
  (void)in_sizes; (void)n_in; (void)out_size; (void)ws_size;

  const float* emb  = (const float*)d_in[0];   // [B, N, D] f32
  const float* W    = (const float*)d_in[1];   // [D] f32
  const float* bias = (const float*)d_in[2];   // scalar f32
  float*       out  = (float*)d_out;           // [B, K, D] f32

  float* scores  = (float*)d_ws;                                   // B*N f32 (512 KB)
  int*   top_idx = (int*)((char*)d_ws + (size_t)BB * NN * sizeof(float)); // B*K i32

  // Pass 1: B*N/16 = 8192 tiles, 8 waves (tiles) per 256-thread block.
  tp_score_wmma<<<(BB * NN / 16) / 8, 256, 0, stream>>>(emb, W, bias, scores);

  // Pass 2: 16 blocks per batch row, 256 threads each.
  tp_rank_topk<<<BB * (NN / 256), 256, 0, stream>>>(scores, top_idx);

  // Pass 3: one block per output row.
  tp_gather<<<BB * KK, 256, 0, stream>>>(emb, top_idx, out);
}